// XGR_82437602279919
// MI455X (gfx1250) — compile-verified
//
#include <hip/hip_runtime.h>
#include <stdint.h>

// ---------------------------------------------------------------------------
// XGR symmetric-Sinkhorn + loss, MI455X / gfx1250
// ---------------------------------------------------------------------------
#define NDIM      2048
#define MAX_ITER  1000
#define WEIGHT    250.0f

#define LOG2E 1.4426950408889634f
#define LN2   0.6931471805599453f

typedef float v2f __attribute__((ext_vector_type(2)));
typedef float v8f __attribute__((ext_vector_type(8)));

// ---- Sinkhorn iteration kernel geometry ----
#define ROWS_PER_WG   16                       // one 16-row strip per workgroup
#define WAVES_PER_WG  8                        // 256 threads (8 x wave32)
#define WG_THREADS    (WAVES_PER_WG * 32)
#define COLS_PER_WAVE (NDIM / WAVES_PER_WG)    // 256 columns per wave
#define TILE_COLS     32                       // 16x32 f32 tile staged in LDS
#define TILES_PER_WAVE (COLS_PER_WAVE / TILE_COLS)  // 8
#define PAD           36                       // dwords per LDS tile row (bank-conflict-free)

// f_next[i] = 0.5*(f[i] - eps*log( sum_j exp((f[j]-M[i,j])/eps) ))
__global__ __launch_bounds__(WG_THREADS)
void xgr_sinkhorn_iter(const float* __restrict__ M,
                       const float* __restrict__ fsrc,
                       float* __restrict__ fdst,
                       const float* __restrict__ eps_p)
{
    __shared__ float f_lds[NDIM];                                  // 8 KB
    __shared__ float tiles[WAVES_PER_WG][2][ROWS_PER_WG * PAD];    // 36 KB, double-buffered
    __shared__ float wave_sums[WAVES_PER_WG][ROWS_PER_WG];         // 0.5 KB

    const int tid      = threadIdx.x;
    const int lane     = tid & 31;
    const int wave     = tid >> 5;
    const int row_base = blockIdx.x * ROWS_PER_WG;

    const float eps = *eps_p;
    const float c1  = LOG2E / eps;          // exp(x) = exp2(x * log2(e))

    // Stage f into LDS (broadcast-read later by all waves).
    for (int i = tid; i < NDIM; i += WG_THREADS) f_lds[i] = fsrc[i];
    __syncthreads();

    const int cbase = wave * COLS_PER_WAVE;

    // Per-lane geometry for async tile copies: each instruction moves
    // 32 lanes x 16 B = 4 rows x 128 B; 4 instructions cover a 16x32 tile.
    const int rsub = lane >> 3;            // 0..3   (row within 4-row group)
    const int csub = (lane & 7) * 4;       // 0..28  (float index within row)
    const uint32_t g_lane =
        (uint32_t)(((row_base + rsub) * NDIM + csub) * (int)sizeof(float));

    auto issue_tile = [&](int bufsel, int c0) {
        const uint32_t lds_base =
            (uint32_t)(size_t)(&tiles[wave][bufsel][0]) +
            (uint32_t)((rsub * PAD + csub) * (int)sizeof(float));
        const uint32_t g_base = g_lane + (uint32_t)(c0 * (int)sizeof(float));
#pragma unroll
        for (int r0 = 0; r0 < ROWS_PER_WG; r0 += 4) {
            uint32_t lds_addr = lds_base + (uint32_t)(r0 * PAD * (int)sizeof(float));
            uint32_t goff     = g_base   + (uint32_t)(r0 * NDIM * (int)sizeof(float));
            // ASYNCcnt-tracked global -> LDS copy (CDNA5).
            asm volatile("global_load_async_to_lds_b128 %0, %1, %2"
                         :: "v"(lds_addr), "v"(goff), "s"(M)
                         : "memory");
        }
    };

    // WMMA operand mapping for V_WMMA_F32_16X16X4_F32:
    //   A (16x4): VGPR0 holds K=0 (lanes 0-15) / K=2 (lanes 16-31), VGPR1 K=1/K=3.
    const int mrow = lane & 15;
    const int ksub = (lane >> 4) * 2;

    v2f ones = {1.0f, 1.0f};               // B = all-ones -> D columns = row sums
    v8f acc  = {0.f, 0.f, 0.f, 0.f, 0.f, 0.f, 0.f, 0.f};

    auto compute_tile = [&](int bufsel, int c0, v8f a_in) -> v8f {
        const float* buf = &tiles[wave][bufsel][0];
        v8f a_acc = a_in;
#pragma unroll
        for (int k = 0; k < TILE_COLS; k += 4) {
            const int col = k + ksub;
            float m0 = buf[mrow * PAD + col];       // conflict-free b64 DS read
            float m1 = buf[mrow * PAD + col + 1];
            float fv0 = f_lds[c0 + col];            // broadcast DS read
            float fv1 = f_lds[c0 + col + 1];
            v2f a;
            a.x = __builtin_amdgcn_exp2f((fv0 - m0) * c1);
            a.y = __builtin_amdgcn_exp2f((fv1 - m1) * c1);
            // D = A x ones + C : accumulates exact f32 row sums on the matrix pipe.
            a_acc = __builtin_amdgcn_wmma_f32_16x16x4_f32(
                        false, a, false, ones, (short)0, a_acc, false, false);
        }
        return a_acc;
    };

    // Double-buffered pipeline: copy tile t+1 while computing tile t.
    issue_tile(0, cbase);
    for (int t = 0; t < TILES_PER_WAVE; ++t) {
        const int sel = t & 1;
        if (t + 1 < TILES_PER_WAVE) {
            issue_tile(sel ^ 1, cbase + (t + 1) * TILE_COLS);
            // 8 in flight, wait until <=4: current tile done (in-order completion).
            asm volatile("s_wait_asynccnt 4" ::: "memory");
        } else {
            asm volatile("s_wait_asynccnt 0" ::: "memory");
        }
        acc = compute_tile(sel, cbase + t * TILE_COLS, acc);
    }

    // D layout: VGPR v / lane 0  -> (M=v,   N=0);  VGPR v / lane 16 -> (M=v+8, N=0).
    if (lane == 0) {
#pragma unroll
        for (int v = 0; v < 8; ++v) wave_sums[wave][v] = acc[v];
    }
    if (lane == 16) {
#pragma unroll
        for (int v = 0; v < 8; ++v) wave_sums[wave][8 + v] = acc[v];
    }
    __syncthreads();

    // Combine the 8 column-partials per row and apply the f update.
    if (tid < ROWS_PER_WG) {
        float S = 0.f;
#pragma unroll
        for (int w = 0; w < WAVES_PER_WG; ++w) S += wave_sums[w][tid];
        const float lse = LN2 * __builtin_amdgcn_logf(S);   // ln(S)
        const int row = row_base + tid;
        fdst[row] = 0.5f * (f_lds[row] - eps * lse);
    }
}

// ---- loss = sum_ij [ g*(ln g - logQ - 1) + exp(logQ) ],  logQ=(f_i+f_j-M_ij)/eps
#define LB_THREADS 256
#define LB_ROWS    (NDIM / 256)    // 8 rows per block, 256 blocks

__global__ __launch_bounds__(LB_THREADS)
void xgr_loss_partial(const float* __restrict__ M,
                      const float* __restrict__ G,
                      const float* __restrict__ f,
                      const float* __restrict__ eps_p,
                      float* __restrict__ partials)
{
    __shared__ float f_lds[NDIM];
    __shared__ float red[LB_THREADS];

    const int tid = threadIdx.x;
    const float eps = *eps_p;
    const float inv_eps = 1.0f / eps;

    for (int i = tid; i < NDIM; i += LB_THREADS) f_lds[i] = f[i];
    __syncthreads();

    float s0 = 0.f, s1 = 0.f, s2 = 0.f, s3 = 0.f;
    const int r0 = blockIdx.x * LB_ROWS;
    for (int r = 0; r < LB_ROWS; ++r) {
        const int row = r0 + r;
        const float fi = f_lds[row];
        const float4* M4 = (const float4*)(M + (size_t)row * NDIM);
        const float4* G4 = (const float4*)(G + (size_t)row * NDIM);
        for (int c4 = tid; c4 < NDIM / 4; c4 += LB_THREADS) {
            const float4 m = M4[c4];
            const float4 g = G4[c4];
            const int c = c4 * 4;
            {
                float lq = (fi + f_lds[c + 0] - m.x) * inv_eps;
                s0 += g.x * (LN2 * __builtin_amdgcn_logf(g.x) - lq - 1.0f)
                    + __builtin_amdgcn_exp2f(lq * LOG2E);
            }
            {
                float lq = (fi + f_lds[c + 1] - m.y) * inv_eps;
                s1 += g.y * (LN2 * __builtin_amdgcn_logf(g.y) - lq - 1.0f)
                    + __builtin_amdgcn_exp2f(lq * LOG2E);
            }
            {
                float lq = (fi + f_lds[c + 2] - m.z) * inv_eps;
                s2 += g.z * (LN2 * __builtin_amdgcn_logf(g.z) - lq - 1.0f)
                    + __builtin_amdgcn_exp2f(lq * LOG2E);
            }
            {
                float lq = (fi + f_lds[c + 3] - m.w) * inv_eps;
                s3 += g.w * (LN2 * __builtin_amdgcn_logf(g.w) - lq - 1.0f)
                    + __builtin_amdgcn_exp2f(lq * LOG2E);
            }
        }
    }

    red[tid] = (s0 + s1) + (s2 + s3);
    __syncthreads();
    for (int s = LB_THREADS / 2; s > 0; s >>= 1) {      // fixed-order tree: deterministic
        if (tid < s) red[tid] += red[tid + s];
        __syncthreads();
    }
    if (tid == 0) partials[blockIdx.x] = red[0];
}

__global__ __launch_bounds__(256)
void xgr_loss_final(const float* __restrict__ partials, float* __restrict__ out)
{
    __shared__ float red[256];
    const int tid = threadIdx.x;
    red[tid] = partials[tid];
    __syncthreads();
    for (int s = 128; s > 0; s >>= 1) {
        if (tid < s) red[tid] += red[tid + s];
        __syncthreads();
    }
    if (tid == 0) out[0] = WEIGHT * red[0];
}

__global__ void xgr_zero(float* __restrict__ p, int n)
{
    int i = blockIdx.x * blockDim.x + threadIdx.x;
    if (i < n) p[i] = 0.f;
}

// ---------------------------------------------------------------------------
extern "C" void kernel_launch(void* const* d_in, const int* in_sizes, int n_in,
                              void* d_out, int out_size, void* d_ws, size_t ws_size,
                              hipStream_t stream)
{
    (void)in_sizes; (void)n_in; (void)out_size; (void)ws_size;

    const float* M     = (const float*)d_in[0];
    const float* G     = (const float*)d_in[1];
    const float* eps_p = (const float*)d_in[2];
    float* out = (float*)d_out;

    float* ws       = (float*)d_ws;
    float* f_a      = ws;              // [0, 2048)
    float* f_b      = ws + NDIM;       // [2048, 4096)
    float* partials = ws + 2 * NDIM;   // [4096, 4352)

    xgr_zero<<<(NDIM + 255) / 256, 256, 0, stream>>>(f_a, NDIM);

    // Fixed-point iterations, ping-pong f buffers. MAX_ITER is even, so the
    // final write (it = 999, odd) lands in f_a.
    for (int it = 0; it < MAX_ITER; ++it) {
        const float* src = (it & 1) ? f_b : f_a;
        float*       dst = (it & 1) ? f_a : f_b;
        xgr_sinkhorn_iter<<<NDIM / ROWS_PER_WG, WG_THREADS, 0, stream>>>(
            M, src, dst, eps_p);
    }

    xgr_loss_partial<<<256, LB_THREADS, 0, stream>>>(M, G, f_a, eps_p, partials);
    xgr_loss_final<<<1, 256, 0, stream>>>(partials, out);
}